// RPN_to_RoI_12068858102122
// MI455X (gfx1250) — compile-verified
//
#include <hip/hip_runtime.h>
#include <hip/hip_bf16.h>
#include <stdint.h>

typedef unsigned long long u64;
typedef unsigned int u32;

#define NB_BATCH 32
#define NH 192
#define NW 192
#define NK 9
#define HWK (NH*NW*NK)          // 331776 = 1024 * 324
#define N_MOS 100
#define NTOP 2048
#define CCAP 4096
#define NBKT 1664
#define BITS_LO 0x3F666667u     // smallest f32 bit pattern strictly > 0.9f
#define SCORE_TH 0.9f
#define IOU_TH 0.9f

// workspace layout (u32 units for the zeroed prefix)
#define HIST_U32 0
#define TB_U32   (NB_BATCH*NBKT)          // 53248
#define CNT_U32  (TB_U32 + NB_BATCH)      // 53280
#define ZERO_N   (CNT_U32 + NB_BATCH)     // 53312 u32 to zero each launch
#define KEYS_BYTE_OFF ((size_t)ZERO_N * 4)  // 213248, 8B aligned
#define WS_NEEDED (KEYS_BYTE_OFF + (size_t)NB_BATCH * CCAP * 8)

#if defined(__gfx1250__) && __has_builtin(__builtin_amdgcn_global_load_async_to_lds_b64) && __has_builtin(__builtin_amdgcn_s_wait_asynccnt)
#define USE_ASYNC_LDS 1
#else
#define USE_ASYNC_LDS 0
#endif

#if USE_ASYNC_LDS
typedef int v2i_t __attribute__((ext_vector_type(2)));
typedef __attribute__((address_space(1))) v2i_t* gv2i_p;   // pointer to global v2i
typedef __attribute__((address_space(3))) v2i_t* lv2i_p;   // pointer to LDS v2i
#endif

__device__ __forceinline__ u32 score_bucket(u32 bits) {
    u32 bk = (bits - BITS_LO) >> 10;      // max 1638 < NBKT
    return bk < NBKT ? bk : (NBKT - 1);
}

// ---------------- kernel 0: zero histograms + counters ----------------
__global__ void zero_ws_kernel(u32* __restrict__ ws) {
    int i = blockIdx.x * blockDim.x + threadIdx.x;
    if (i < ZERO_N) ws[i] = 0u;
}

// ---------------- kernel 1: per-batch score-bit histogram ----------------
__global__ __launch_bounds__(256) void hist_kernel(const float* __restrict__ score,
                                                   u32* __restrict__ ws) {
    __shared__ u32 lh[NBKT];
    const int b = blockIdx.y;
    for (int i = threadIdx.x; i < NBKT; i += 256) lh[i] = 0u;
    __syncthreads();

    const float4* sp4 = (const float4*)(score + (size_t)b * HWK + (size_t)blockIdx.x * 1024);
    float4 v = sp4[threadIdx.x];
    float sv[4] = {v.x, v.y, v.z, v.w};
#pragma unroll
    for (int c = 0; c < 4; ++c) {
        if (sv[c] > SCORE_TH) {
            atomicAdd(&lh[score_bucket(__float_as_uint(sv[c]))], 1u);
        }
    }
    __syncthreads();
    u32* gh = ws + HIST_U32 + (size_t)b * NBKT;
    for (int i = threadIdx.x; i < NBKT; i += 256) {
        u32 cnt = lh[i];
        if (cnt) atomicAdd(&gh[i], cnt);
    }
}

// ---------------- kernel 2: find threshold bucket (rank 2048) ----------------
__global__ void thresh_kernel(u32* __restrict__ ws) {
    __shared__ u32 lh[NBKT];
    const int b = blockIdx.x;
    const u32* gh = ws + HIST_U32 + (size_t)b * NBKT;
    for (int i = threadIdx.x; i < NBKT; i += blockDim.x) lh[i] = gh[i];
    __syncthreads();
    if (threadIdx.x == 0) {
        u32 acc = 0;
        int tb = 0;
        for (int i = NBKT - 1; i >= 0; --i) {
            acc += lh[i];
            if (acc >= NTOP) { tb = i; break; }
        }
        ws[TB_U32 + b] = (u32)tb;
    }
}

// ---------------- kernel 3: compact candidate keys >= threshold bucket ----------------
__global__ __launch_bounds__(256) void compact_kernel(const float* __restrict__ score,
                                                      u32* __restrict__ ws) {
    __shared__ u32 s_lcnt, s_gbase;
    const int b = blockIdx.y;
    const u32 tb = ws[TB_U32 + b];
    u64* keys = (u64*)((char*)ws + KEYS_BYTE_OFF) + (size_t)b * CCAP;

    if (threadIdx.x == 0) s_lcnt = 0u;
    __syncthreads();

    const u32 base_idx = blockIdx.x * 1024u + threadIdx.x * 4u;
    const float4* sp4 = (const float4*)(score + (size_t)b * HWK + (size_t)blockIdx.x * 1024);
    float4 v = sp4[threadIdx.x];
    float sv[4] = {v.x, v.y, v.z, v.w};

    bool ok[4];
    u32 lpos[4];
    u32 bits[4];
#pragma unroll
    for (int c = 0; c < 4; ++c) {
        ok[c] = false;
        if (sv[c] > SCORE_TH) {
            u32 bt = __float_as_uint(sv[c]);
            if (score_bucket(bt) >= tb) {
                bits[c] = bt;
                lpos[c] = atomicAdd(&s_lcnt, 1u);
                ok[c] = true;
            }
        }
    }
    __syncthreads();
    if (threadIdx.x == 0) s_gbase = atomicAdd(ws + CNT_U32 + b, s_lcnt);
    __syncthreads();
    const u32 gbase = s_gbase;
#pragma unroll
    for (int c = 0; c < 4; ++c) {
        if (ok[c]) {
            u32 p = gbase + lpos[c];
            if (p < CCAP) {
                u32 idx = base_idx + (u32)c;
                keys[p] = ((u64)bits[c] << 32) | (u32)(~idx);
            }
        }
    }
}

// ---------------- kernel 4: per-batch exact top-2048 + serial NMS ----------------
__global__ __launch_bounds__(1024, 1) void nms_kernel(const float* __restrict__ delta,
                                                      const float* __restrict__ anch,
                                                      const u32* __restrict__ ws,
                                                      float* __restrict__ out) {
    __shared__ u64 keys[CCAP];            // 32 KB; upper half reused for by2/bx1 after sort
    __shared__ float by1[NTOP];           // 8 KB
    __shared__ float bx2[NTOP];           // 8 KB
    __shared__ u64 wkey[32];
    __shared__ int wslot[32];
    __shared__ u64 s_wk;
    __shared__ int s_ws;

    const int b = blockIdx.x;
    const int t = threadIdx.x;

    u32 C = ws[CNT_U32 + b];
    if (C > CCAP) C = CCAP;
    const u64* gkeys = (const u64*)((const char*)ws + KEYS_BYTE_OFF) + (size_t)b * CCAP;

    // ---- stage candidate keys into LDS (async-to-LDS path on gfx1250) ----
    for (int i = t; i < CCAP; i += 1024) {
        if (i < (int)C) {
#if USE_ASYNC_LDS
            __builtin_amdgcn_global_load_async_to_lds_b64(
                (gv2i_p)(gkeys + i), (lv2i_p)(&keys[i]), 0, 0);
#else
            keys[i] = gkeys[i];
#endif
        } else {
            keys[i] = 0ull;
        }
    }
#if USE_ASYNC_LDS
    __builtin_amdgcn_s_wait_asynccnt(0);
#endif
    __syncthreads();

    // ---- bitonic sort 4096 keys, descending => keys[0..2047] = exact top-2048 set ----
    for (int k = 2; k <= CCAP; k <<= 1) {
        for (int j = k >> 1; j > 0; j >>= 1) {
            for (int i = t; i < CCAP; i += 1024) {
                int ixj = i ^ j;
                if (ixj > i) {
                    bool up = ((i & k) == 0);
                    u64 a = keys[i], c2 = keys[ixj];
                    bool sw = up ? (a < c2) : (a > c2);
                    if (sw) { keys[i] = c2; keys[ixj] = a; }
                }
            }
            __syncthreads();
        }
    }

    // upper half of sort buffer is now free: reuse as box storage
    float* by2 = (float*)&keys[NTOP];     // 8 KB
    float* bx1 = by2 + NTOP;              // 8 KB

    // ---- gather deltas/anchors for the 2048 survivors and decode boxes ----
    for (int s = t; s < NTOP; s += 1024) {
        u64 kk = keys[s];
        float oy2 = 0.f, ox1 = 0.f, oy1 = 0.f, ox2 = 0.f;
        if (kk) {
            u32 idx = ~(u32)(kk & 0xFFFFFFFFull);
            const float* ap = anch + (size_t)idx * 4;
            const float* dp = delta + ((size_t)b * HWK + idx) * 4;
            float a0 = ap[0], a1 = ap[1], a2 = ap[2], a3 = ap[3];
            float tx = dp[0], ty = dp[1], tw = dp[2], th = dp[3];
            float xa = (a0 + a1) * 0.5f, ya = (a2 + a3) * 0.5f;
            float wa = a1 - a0, ha = a3 - a2;
            float x = tx * wa + xa, y = ty * ha + ya;
            float w = expf(tw) * wa, h = expf(th) * ha;
            ox1 = fmaxf(x - w * 0.5f, 0.f);
            ox2 = fminf(x + w * 0.5f, 1.f);
            oy1 = fmaxf(y - h * 0.5f, 0.f);
            oy2 = fminf(y + h * 0.5f, 1.f);
        }
        by2[s] = oy2; bx1[s] = ox1; by1[s] = oy1; bx2[s] = ox2;
    }
    __syncthreads();

    const int lane = t & 31;
    const int wid = t >> 5;

    // ---- 100-step serial NMS ----
    for (int it = 0; it < N_MOS; ++it) {
        // two-level argmax over keys[0..2047] (key order == score order, tie -> lower index)
        u64 bk = 0ull;
        int bs = 0;
        {
            u64 k0 = keys[t];        if (k0 > bk) { bk = k0; bs = t; }
            u64 k1 = keys[t + 1024]; if (k1 > bk) { bk = k1; bs = t + 1024; }
        }
#pragma unroll
        for (int off = 16; off > 0; off >>= 1) {
            u64 ok = __shfl_xor(bk, off);
            int os = __shfl_xor(bs, off);
            if (ok > bk) { bk = ok; bs = os; }
        }
        if (lane == 0) { wkey[wid] = bk; wslot[wid] = bs; }
        __syncthreads();
        if (t < 32) {
            bk = wkey[t]; bs = wslot[t];
#pragma unroll
            for (int off = 16; off > 0; off >>= 1) {
                u64 ok = __shfl_xor(bk, off);
                int os = __shfl_xor(bs, off);
                if (ok > bk) { bk = ok; bs = os; }
            }
            if (t == 0) { s_wk = bk; s_ws = bs; }
        }
        __syncthreads();

        const u64 W = s_wk;
        const int wsl = s_ws;
        const bool valid = (W != 0ull);

        float wy2 = 0.f, wx1 = 0.f, wy1 = 0.f, wx2 = 0.f;
        if (valid) { wy2 = by2[wsl]; wx1 = bx1[wsl]; wy1 = by1[wsl]; wx2 = bx2[wsl]; }

        if (t < 4) {
            float v = (t == 0) ? wy2 : (t == 1) ? wx1 : (t == 2) ? wy1 : wx2;
            out[(size_t)b * N_MOS * 4 + (size_t)it * 4 + t] = valid ? v : 0.f;
        }

        if (valid) {
            float area_a = fmaxf(wy2 - wy1, 0.f) * fmaxf(wx2 - wx1, 0.f);
#pragma unroll
            for (int r = 0; r < 2; ++r) {
                int s = t + r * 1024;
                if (keys[s] != 0ull) {
                    float y2b = by2[s], x1b = bx1[s], y1b = by1[s], x2b = bx2[s];
                    float area_b = fmaxf(y2b - y1b, 0.f) * fmaxf(x2b - x1b, 0.f);
                    float ih = fmaxf(fminf(wy2, y2b) - fmaxf(wy1, y1b), 0.f);
                    float iw = fmaxf(fminf(wx2, x2b) - fmaxf(wx1, x1b), 0.f);
                    float inter = ih * iw;
                    float iou = inter / (area_a + area_b - inter + 1e-9f);
                    if (iou > IOU_TH) keys[s] = 0ull;
                }
            }
            if (t == 0) keys[wsl] = 0ull;   // reference explicitly kills the selected entry
        }
        __syncthreads();
    }
}

extern "C" void kernel_launch(void* const* d_in, const int* in_sizes, int n_in,
                              void* d_out, int out_size, void* d_ws, size_t ws_size,
                              hipStream_t stream) {
    (void)in_sizes; (void)n_in; (void)out_size;
    if (ws_size < WS_NEEDED) return;   // cannot run without scratch

    const float* score = (const float*)d_in[0];   // (32,192,192,9) f32
    const float* delta = (const float*)d_in[1];   // (32,192,192,36) f32
    const float* anch  = (const float*)d_in[2];   // (192,192,9,4) f32
    float* out = (float*)d_out;                   // (32,100,4) f32
    u32* ws = (u32*)d_ws;

    zero_ws_kernel<<<(ZERO_N + 255) / 256, 256, 0, stream>>>(ws);

    dim3 g(HWK / 1024, NB_BATCH);   // (324, 32)
    hist_kernel<<<g, 256, 0, stream>>>(score, ws);
    thresh_kernel<<<NB_BATCH, 256, 0, stream>>>(ws);
    compact_kernel<<<g, 256, 0, stream>>>(score, ws);
    nms_kernel<<<NB_BATCH, 1024, 0, stream>>>(delta, anch, ws, out);
}